// WHT1D_85280870629431
// MI455X (gfx1250) — compile-verified
//
#include <hip/hip_runtime.h>
#include <hip/hip_bf16.h>

// ---------------------------------------------------------------------------
// WHT-512 gate/soft-threshold pipeline, factored as H512 = H2 (x) H16 (x) H16.
// Each wave processes one row of 512 fp32 values:
//   load -> [per half: H16 * X * H16 via two chained f32-WMMA passes with an
//   LDS transpose bounce] -> H2 butterfly + scale + v-gate + soft-threshold ->
//   same inverse structure -> butterfly + scale -> +residual -> store.
// All matrix math uses V_WMMA_F32_16X16X4_F32 (exact fp32, K chained x4).
// The Hadamard operand is synthesized in registers: H[m][k] = (-1)^popc(m&k).
// ---------------------------------------------------------------------------

typedef float v2f __attribute__((ext_vector_type(2)));
typedef float v8f __attribute__((ext_vector_type(8)));

#define WAVES_PER_BLOCK 8
#define THREADS (WAVES_PER_BLOCK * 32)
#define RS 20            // padded row stride (floats) of a 16x16 tile (16B-aligned rows)
#define MAT (16 * RS)    // 320 floats per padded tile
#define XS_FLOATS (2 * MAT)

__device__ __forceinline__ float hsign(int m, int k) {
  return (__popc(m & k) & 1) ? -1.0f : 1.0f;
}

// D = H16 * S  (or H16 * S^T when `transposed`), S is a padded 16x16 LDS tile.
// A operand = H16 in native A layout (VGPR j of chunk kb holds K = 4kb+2*half+j,
// row m = lane&15). B operand mirrors that K placement with N = lane&15.
// Result is in C/D layout: acc[i] = D[i + 8*half][lane&15].
__device__ __forceinline__ v8f hmul16(const float* __restrict__ S, int mn, int half,
                                      const v2f* Ha, bool transposed) {
  v8f acc = {0.f, 0.f, 0.f, 0.f, 0.f, 0.f, 0.f, 0.f};
#pragma unroll
  for (int kb = 0; kb < 4; ++kb) {
    const int k0 = 4 * kb + 2 * half;
    v2f b;
    if (!transposed) {
      b.x = S[k0 * RS + mn];
      b.y = S[(k0 + 1) * RS + mn];
    } else {
      b.x = S[mn * RS + k0];
      b.y = S[mn * RS + k0 + 1];
    }
    acc = __builtin_amdgcn_wmma_f32_16x16x4_f32(
        /*neg_a=*/false, Ha[kb], /*neg_b=*/false, b,
        /*c_mod=*/(short)0, acc, /*reuse_a=*/false, /*reuse_b=*/false);
  }
  return acc;
}

__global__ __launch_bounds__(THREADS)
void wht512_fused_kernel(const float* __restrict__ x, const float* __restrict__ v,
                         const float* __restrict__ T, float* __restrict__ out,
                         int rows) {
  __shared__ float sV[XS_FLOATS];
  __shared__ float sT[XS_FLOATS];
  __shared__ float sX[WAVES_PER_BLOCK][XS_FLOATS];  // X -> W -> final, reused
  __shared__ float sU[WAVES_PER_BLOCK][MAT];        // transpose bounce buffer

  const int tid = threadIdx.x;
  // Preload v, T once per block into the padded (a, r, c) layout.
  for (int e = tid; e < 512; e += THREADS) {
    const int a = e >> 8, r = (e >> 4) & 15, c = e & 15;
    const int off = a * MAT + r * RS + c;
    sV[off] = v[e];
    sT[off] = T[e];
  }
  __syncthreads();

  const int lane = tid & 31;
  const int wave = tid >> 5;
  const int mn = lane & 15;   // row index m for A operand / col index n for B,D
  const int half = lane >> 4;

  v2f Ha[4];
#pragma unroll
  for (int kb = 0; kb < 4; ++kb) {
    const int k0 = 4 * kb + 2 * half;
    Ha[kb].x = hsign(mn, k0);
    Ha[kb].y = hsign(mn, k0 + 1);
  }

  float* xs = sX[wave];
  float* us = sU[wave];
  const float scale = 0.044194173824159216f;  // 1/sqrt(512)

  const long long gwave = (long long)blockIdx.x * WAVES_PER_BLOCK + wave;
  const long long nwaves = (long long)gridDim.x * WAVES_PER_BLOCK;
  const int a0 = lane >> 4, r0 = lane & 15;   // this lane's contiguous segment

  for (long long row = gwave; row < rows; row += nwaves) {
    const float* xr = x + (size_t)row * 512;
    float* orow = out + (size_t)row * 512;

    // -- coalesced load: lane owns 16 contiguous floats; keep as residual ----
    float4 xv[4];
#pragma unroll
    for (int q = 0; q < 4; ++q)
      xv[q] = ((const float4*)xr)[lane * 4 + q];
#pragma unroll
    for (int q = 0; q < 4; ++q)
      *(float4*)&xs[a0 * MAT + r0 * RS + 4 * q] = xv[q];
    asm volatile("" ::: "memory");

    // -- forward: Y_a = H16 * X_a * H16 (two WMMA passes, LDS transpose) -----
    v8f y[2];
#pragma unroll
    for (int a = 0; a < 2; ++a) {
      v8f u = hmul16(xs + a * MAT, mn, half, Ha, false);      // U = H * X_a
#pragma unroll
      for (int i = 0; i < 8; ++i)
        us[(i + 8 * half) * RS + mn] = u[i];
      asm volatile("" ::: "memory");
      y[a] = hmul16(us, mn, half, Ha, true);                  // (H*X*H)^T
      asm volatile("" ::: "memory");
    }
    // y[a][i] = Y_a[n][m] with n = mn, m = i + 8*half.

    // -- H2 butterfly + scale + gate + soft-threshold; store W_a^T into xs ---
#pragma unroll
    for (int i = 0; i < 8; ++i) {
      const int m = i + 8 * half;
      const int off = mn * RS + m;          // feature (a, r=n, c=m), padded
      const float f0 = (y[0][i] + y[1][i]) * scale;
      const float f1 = (y[0][i] - y[1][i]) * scale;
      const float g0 = sV[off] * f0;
      const float g1 = sV[MAT + off] * f1;
      const float w0 = copysignf(fmaxf(fabsf(g0) - sT[off], 0.0f), g0);
      const float w1 = copysignf(fmaxf(fabsf(g1) - sT[MAT + off], 0.0f), g1);
      xs[0 * MAT + m * RS + mn] = w0;       // stores Q = W_a^T (conflict-free)
      xs[1 * MAT + m * RS + mn] = w1;
    }
    asm volatile("" ::: "memory");

    // -- inverse: Z_a = H16 * W_a * H16 (identical structure; transposes
    //    cancel because H is symmetric) -----------------------------------
    v8f z[2];
#pragma unroll
    for (int a = 0; a < 2; ++a) {
      v8f u = hmul16(xs + a * MAT, mn, half, Ha, false);      // H * W_a^T
#pragma unroll
      for (int i = 0; i < 8; ++i)
        us[(i + 8 * half) * RS + mn] = u[i];
      asm volatile("" ::: "memory");
      z[a] = hmul16(us, mn, half, Ha, true);                  // Z_a[m][n]
      asm volatile("" ::: "memory");
    }

    // -- final butterfly + scale -> natural padded layout in xs --------------
#pragma unroll
    for (int i = 0; i < 8; ++i) {
      const int m = i + 8 * half;
      xs[0 * MAT + m * RS + mn] = (z[0][i] + z[1][i]) * scale;
      xs[1 * MAT + m * RS + mn] = (z[0][i] - z[1][i]) * scale;
    }
    asm volatile("" ::: "memory");

    // -- coalesced store with residual add from registers --------------------
#pragma unroll
    for (int q = 0; q < 4; ++q) {
      const float4 t = *(const float4*)&xs[a0 * MAT + r0 * RS + 4 * q];
      float4 o;
      o.x = t.x + xv[q].x;
      o.y = t.y + xv[q].y;
      o.z = t.z + xv[q].z;
      o.w = t.w + xv[q].w;
      ((float4*)orow)[lane * 4 + q] = o;
    }
  }
}

extern "C" void kernel_launch(void* const* d_in, const int* in_sizes, int n_in,
                              void* d_out, int out_size, void* d_ws, size_t ws_size,
                              hipStream_t stream) {
  const float* x = (const float*)d_in[0];
  const float* v = (const float*)d_in[1];
  const float* T = (const float*)d_in[2];
  float* out = (float*)d_out;

  const int rows = in_sizes[0] / 512;  // (B*L) rows of N=512
  int blocks = (rows + WAVES_PER_BLOCK - 1) / WAVES_PER_BLOCK;
  if (blocks > 4096) blocks = 4096;    // loop inside kernel; saturate HBM
  if (blocks < 1) blocks = 1;

  wht512_fused_kernel<<<blocks, THREADS, 0, stream>>>(x, v, T, out, rows);
}